// MoELIFNode_68186900791885
// MI455X (gfx1250) — compile-verified
//
#include <hip/hip_runtime.h>
#include <hip/hip_bf16.h>
#include <math.h>

// MoE-LIF fused kernel for MI455X (gfx1250, wave32).
// Memory-bound: 64MiB in + 64MiB out @ 23.3 TB/s => ~5.8us floor.
// One block = (batch b, 32 columns of N).
//  - x tile (256 rows x 32 f32, row stride 4096) staged by the Tensor Data
//    Mover (tensor_load_to_lds) with hardware LDS padding (33-dword rows).
//  - gate GEMM (E=4 x K=256) via chained V_WMMA_F32_16X16X4_F32 (exact fp32).
//  - top-2 masked softmax per-lane, expert recurrences + heaviside fused,
//    nontemporal output stores.

typedef __attribute__((ext_vector_type(2))) float v2f;
typedef __attribute__((ext_vector_type(8))) float v8f;
typedef __attribute__((ext_vector_type(4))) unsigned int v4u;
typedef __attribute__((ext_vector_type(4))) int v4i;
typedef __attribute__((ext_vector_type(8))) int v8i;

#define T_STEPS 4
#define B_BATCH 16
#define C_CH    64
#define N_DIM   4096
#define K_TOT   256            // T*C contraction length for the gate
#define NTILE   32             // n columns per block
#define XSTRIDE 33             // padded LDS row stride (TDM pad: +1 dword / 32)
#define VTH     0.2f
#define THETA   0.8f

__device__ __forceinline__ float fire_reset(float& v) {
    // spike = (v >= VTH); v *= (1-spike); ev = (v==0) ? VTH : v
    v = (v >= VTH) ? 0.0f : v;
    return (v == 0.0f) ? VTH : v;
}

__global__ __launch_bounds__(128)
void moe_lif_fused(const float* __restrict__ x,
                   const float* __restrict__ gw,   // (4, 256)
                   const float* __restrict__ gb,   // (4,)
                   const float* __restrict__ pw,   // (1,)
                   float* __restrict__ out) {
    __shared__ __align__(16) float xt_s[K_TOT * XSTRIDE]; // z tile [k=t*64+c][n], 33 KB
    __shared__ float wt_s[K_TOT * 16];        // weights [k][e], rows e>=4 zeroed, 16 KB
    __shared__ float gate_s[NTILE * 4];       // softmaxed top-2 gate weights [n][e]

    const int tid = threadIdx.x;
    const int b   = blockIdx.y;
    const int nb  = blockIdx.x * NTILE;

    // ---- TDM: DMA the 256x32 f32 tile global->LDS with +1-dword row padding ----
    if (tid < 32) {   // wave 0 only (uniform branch; TDM ignores EXEC, 1 issue)
        const unsigned lds_off = (unsigned)(size_t)&xt_s[0];
        const unsigned long long ga =
            (unsigned long long)(size_t)x
            + ((unsigned long long)b * K_TOT * N_DIM + (unsigned)nb) * 4ull;
        // D# group0: count=1 | lds_addr | global_addr[56:0] | type=2
        v4u g0 = { 1u,
                   lds_off,
                   (unsigned)(ga & 0xFFFFFFFFu),
                   (unsigned)((ga >> 32) & 0x01FFFFFFu) | 0x80000000u };
        // D# group1:
        //  d0: data_size=2(4B)<<16 | pad_enable<<20 | pad_interval=4(32dw)<<22
        //      | pad_amount=0(1dw)<<25                      = 0x01120000
        //  d1: tensor_dim0(4096) low16 << 16                = 0x10000000
        //  d2: tensor_dim0 hi=0 | tensor_dim1(256) low16<<16= 0x01000000
        //  d3: tensor_dim1 hi=0 | tile_dim0(32)<<16         = 0x00200000
        //  d4: tile_dim1(256) | tile_dim2(0)<<16            = 0x00000100
        //  d5: tensor_dim0_stride(4096) low32               = 0x00001000
        //  d6,d7: stride hi / tensor_dim1_stride (unused)   = 0
        v8i g1 = { 0x01120000, 0x10000000, 0x01000000, 0x00200000,
                   0x00000100, 0x00001000, 0, 0 };
        v4i g2 = { 0, 0, 0, 0 };      // D# group2 (unused, 2D tensor)
        v4i g3 = { 0, 0, 0, 0 };      // D# group3 (unused, 2D tensor)
        v8i g4 = { 0, 0, 0, 0, 0, 0, 0, 0 };  // extra group (clang-23 6-arg form)
        __builtin_amdgcn_tensor_load_to_lds(g0, g1, g2, g3, g4, 0);
    }

    // ---- stage gate weights (zero-pad expert rows 4..15 for the 16x16 WMMA) ----
    for (int i = tid; i < K_TOT * 16; i += 128) {
        int e = i & 15, k = i >> 4;
        wt_s[i] = (e < 4) ? gw[e * K_TOT + k] : 0.0f;
    }
    __builtin_amdgcn_s_wait_tensorcnt(0);   // no-op for waves 1..3
    __syncthreads();

    // ---- gate GEMM: D(16e x 16n) += A(16e x 4k) * B(4k x 16n), 64 K-steps ----
    const int wv   = tid >> 5;
    const int lane = tid & 31;
    if (wv < 2) {                                  // 2 waves cover 32 n columns
        const int koff = (lane < 16) ? 0 : 2;      // K split across half-waves
        const int l16  = lane & 15;                // A: row e ; B: col n
        const int nB   = wv * 16 + l16;
        v8f acc = {0.f, 0.f, 0.f, 0.f, 0.f, 0.f, 0.f, 0.f};
        #pragma unroll 4
        for (int k0 = 0; k0 < K_TOT; k0 += 4) {
            v2f a, bf;
            a.x  = wt_s[(k0 + koff)     * 16 + l16];
            a.y  = wt_s[(k0 + koff + 1) * 16 + l16];
            bf.x = xt_s[(k0 + koff)     * XSTRIDE + nB];
            bf.y = xt_s[(k0 + koff + 1) * XSTRIDE + nB];
            acc = __builtin_amdgcn_wmma_f32_16x16x4_f32(
                false, a, false, bf, (short)0, acc, false, false);
        }
        // lanes 0..15 hold rows e=0..7 in acc[0..7] for column n = lane
        if (lane < 16) {
            float g[4];
            #pragma unroll
            for (int e = 0; e < 4; e++) g[e] = acc[e] + gb[e];
            // top-2 (JAX tie-break: lowest index wins on ties)
            int i0 = 0;
            #pragma unroll
            for (int e = 1; e < 4; e++) if (g[e] > g[i0]) i0 = e;
            int i1 = (i0 == 0) ? 1 : 0;
            #pragma unroll
            for (int e = 0; e < 4; e++) if (e != i0 && g[e] > g[i1]) i1 = e;
            float v0 = g[i0], v1 = g[i1];           // v0 >= v1
            // masked softmax: selected-but-zero values become -inf (ref quirk)
            float e0 = (v0 != 0.0f) ? 1.0f : 0.0f;
            float e1 = (v1 != 0.0f) ? expf(v1 - v0) : 0.0f;
            float s  = e0 + e1;
            float w0 = e0 / s, w1 = e1 / s;
            float wsel[4] = {0.f, 0.f, 0.f, 0.f};
            wsel[i0] = w0; wsel[i1] = w1;
            const int n_l = wv * 16 + lane;
            gate_s[n_l * 4 + 0] = wsel[0];
            gate_s[n_l * 4 + 1] = wsel[1];
            gate_s[n_l * 4 + 2] = wsel[2];
            gate_s[n_l * 4 + 3] = wsel[3];
        }
    }
    __syncthreads();

    // ---- expert recurrences + mix + heaviside (nontemporal streamed out) ----
    const float sp = 1.0f / (1.0f + expf(-pw[0]));   // sigmoid(plif_w)
    for (int p = tid; p < C_CH * NTILE; p += 128) {
        int c = p >> 5, n = p & (NTILE - 1);
        float g0 = gate_s[n * 4 + 0];
        float g1 = gate_s[n * 4 + 1];
        float g2 = gate_s[n * 4 + 2];
        float g3 = gate_s[n * 4 + 3];
        float vl = 0.f, ve = 0.f, vp = 0.f, vi = 0.f;
        #pragma unroll
        for (int t = 0; t < T_STEPS; t++) {
            float xv = xt_s[(t * 64 + c) * XSTRIDE + n];
            vl = vl + (xv - vl) * 0.5f;                          // LIF, tau=2
            float el = fire_reset(vl);
            ve = ve + (xv - ve + expf(ve - THETA)) * 0.5f;       // EIF
            float ee = fire_reset(ve);
            vp = vp + (xv - vp) * sp;                            // PLIF
            float ep = fire_reset(vp);
            vi = vi + xv;                                        // IF
            float ei = fire_reset(vi);
            float o = g0 * el + g1 * ee + g2 * ep + g3 * ei;
            float r = (o >= VTH) ? 1.0f : 0.0f;
            __builtin_nontemporal_store(
                r, &out[((size_t)(t * B_BATCH + b) * C_CH + c) * N_DIM + nb + n]);
        }
    }
}

extern "C" void kernel_launch(void* const* d_in, const int* in_sizes, int n_in,
                              void* d_out, int out_size, void* d_ws, size_t ws_size,
                              hipStream_t stream) {
    (void)in_sizes; (void)n_in; (void)out_size; (void)d_ws; (void)ws_size;
    const float* x  = (const float*)d_in[0];
    const float* gw = (const float*)d_in[1];
    const float* gb = (const float*)d_in[2];
    const float* pw = (const float*)d_in[3];
    float* out = (float*)d_out;
    dim3 grid(N_DIM / NTILE, B_BATCH);   // 128 n-tiles x 16 batches
    moe_lif_fused<<<grid, 128, 0, stream>>>(x, gw, gb, pw, out);
}